// SKLearnLLE_59433757442766
// MI455X (gfx1250) — compile-verified
//
#include <hip/hip_runtime.h>
#include <hip/hip_bf16.h>

typedef float v2f __attribute__((ext_vector_type(2)));
typedef float v8f __attribute__((ext_vector_type(8)));
typedef int   b128_t __attribute__((vector_size(16)));   // async-LDS payload type

constexpr int N   = 4096;
constexpr int D   = 1024;
constexpr int KNN = 10;
constexpr int P   = 48;    // subspace width (>= 33 needed)
constexpr int NC  = 32;    // output components
constexpr int ITERS = 128; // subspace iterations
constexpr int KSPL  = 8;   // K-split blocks for P x P Gram reductions

#if defined(__has_builtin)
#if __has_builtin(__builtin_amdgcn_global_load_async_to_lds_b128) && \
    __has_builtin(__builtin_amdgcn_s_wait_asynccnt)
#define USE_ASYNC_LDS 1
#endif
#endif

static __device__ __forceinline__ v8f wmma4(v2f a, v2f b, v8f c) {
  // V_WMMA_F32_16X16X4_F32 : D = A(16x4) * B(4x16) + C(16x16)
  return __builtin_amdgcn_wmma_f32_16x16x4_f32(false, a, false, b, (short)0, c,
                                               false, false);
}

static __device__ __forceinline__ float wave_sum(float s) {
  for (int o = 16; o; o >>= 1) s += __shfl_xor(s, o, 32);
  return s;
}

// ---------------- row squared norms ----------------
__global__ void k_rownorm(const float* __restrict__ x, float* __restrict__ sq) {
  int row = blockIdx.x * 8 + (threadIdx.x >> 5);
  int lane = threadIdx.x & 31;
  float s = 0.f;
  const float* r = x + (size_t)row * D;
  for (int d = lane; d < D; d += 32) { float v = r[d]; s += v * v; }
  s = wave_sum(s);
  if (lane == 0) sq[row] = s;
}

// ---------------- d2 = |xi|^2 + |xj|^2 - 2 x x^T  (f32 WMMA) ----------------
__global__ void k_dist(const float* __restrict__ x, const float* __restrict__ sq,
                       float* __restrict__ d2) {
  const int lane = threadIdx.x & 31;
  const int wave = threadIdx.x >> 5;
  const int rowBase = blockIdx.y * 128 + wave * 16;
  const int colBase = blockIdx.x * 64;
  const int m  = lane & 15;
  const int kb = (lane >> 4) * 2;

  v8f accA[4], accB[4];
  v8f z = {0.f,0.f,0.f,0.f,0.f,0.f,0.f,0.f};
  for (int t = 0; t < 4; ++t) { accA[t] = z; accB[t] = z; }

  const float* arow = x + (size_t)(rowBase + m) * D + kb;
  for (int k0 = 0; k0 < D; k0 += 8) {
    __builtin_prefetch(arow + k0 + 256, 0, 1);   // global_prefetch_b8
    v2f a0 = *(const v2f*)(arow + k0);
    v2f a1 = *(const v2f*)(arow + k0 + 4);
#pragma unroll
    for (int t = 0; t < 4; ++t) {
      int n = colBase + t * 16 + m;
      const float* brow = x + (size_t)n * D + k0 + kb;
      v2f b0 = *(const v2f*)(brow);
      v2f b1 = *(const v2f*)(brow + 4);
      accA[t] = wmma4(a0, b0, accA[t]);
      accB[t] = wmma4(a1, b1, accB[t]);
    }
  }
  const int mb = rowBase + (lane >> 4) * 8;
#pragma unroll
  for (int t = 0; t < 4; ++t) {
    v8f acc = accA[t] + accB[t];
    int n = colBase + t * 16 + m;
    float sqn = sq[n];
#pragma unroll
    for (int r = 0; r < 8; ++r) {
      int i = mb + r;
      float v = sq[i] + sqn - 2.0f * acc[r];
      if (i == n) v += 1e10f;
      d2[(size_t)i * N + n] = v;
    }
  }
}

// ---------------- top-k smallest per row ----------------
__global__ void k_topk(const float* __restrict__ d2, int* __restrict__ idx) {
  __shared__ float sv[32 * KNN];
  __shared__ int   si[32 * KNN];
  int row = blockIdx.x;
  int lane = threadIdx.x & 31;
  float bv[KNN]; int bi[KNN];
  for (int t = 0; t < KNN; ++t) { bv[t] = 3.0e38f; bi[t] = 0; }
  const float* r = d2 + (size_t)row * N;
  for (int j = lane; j < N; j += 32) {
    float v = r[j];
    if (v < bv[KNN - 1]) {
      int p = KNN - 1;
      while (p > 0 && bv[p - 1] > v) { bv[p] = bv[p - 1]; bi[p] = bi[p - 1]; --p; }
      bv[p] = v; bi[p] = j;
    }
  }
  for (int t = 0; t < KNN; ++t) { sv[lane * KNN + t] = bv[t]; si[lane * KNN + t] = bi[t]; }
  __syncthreads();
  if (lane == 0) {
    for (int t = 0; t < KNN; ++t) {
      float best = 3.0e38f; int bj = 0;
      for (int q = 0; q < 32 * KNN; ++q)
        if (sv[q] < best) { best = sv[q]; bj = q; }
      idx[row * KNN + t] = si[bj];
      sv[bj] = 3.3e38f;
    }
  }
}

// ---------------- barycentric weights (10x10 regularized solve) ----------------
__global__ void k_weights(const float* __restrict__ x, const int* __restrict__ idx,
                          float* __restrict__ w) {
  __shared__ float Zs[KNN * D];      // 40 KB of the 320 KB WGP LDS pool
  __shared__ float Gsh[KNN][KNN];
  int row = blockIdx.x;
  int lane = threadIdx.x & 31;
  int nb[KNN];
  for (int t = 0; t < KNN; ++t) nb[t] = idx[row * KNN + t];
  const float* xi = x + (size_t)row * D;
  for (int a = 0; a < KNN; ++a) {
    const float* xa = x + (size_t)nb[a] * D;
    for (int d = lane; d < D; d += 32) Zs[a * D + d] = xa[d] - xi[d];
  }
  __syncthreads();
  for (int a = 0; a < KNN; ++a)
    for (int b = a; b < KNN; ++b) {
      float s = 0.f;
      for (int d = lane; d < D; d += 32) s += Zs[a * D + d] * Zs[b * D + d];
      s = wave_sum(s);
      if (lane == 0) { Gsh[a][b] = s; Gsh[b][a] = s; }
    }
  __syncthreads();
  if (lane == 0) {
    float A[KNN][KNN], L[KNN][KNN];
    float tr = 0.f;
    for (int t = 0; t < KNN; ++t) tr += Gsh[t][t];
    for (int a = 0; a < KNN; ++a)
      for (int b = 0; b < KNN; ++b) A[a][b] = Gsh[a][b];
    for (int t = 0; t < KNN; ++t) A[t][t] += 1e-3f * tr;   // REG * trace
    for (int c = 0; c < KNN; ++c) {
      float s = A[c][c];
      for (int j = 0; j < c; ++j) s -= L[c][j] * L[c][j];
      float dc = sqrtf(fmaxf(s, 1e-20f));
      L[c][c] = dc;
      for (int rr = c + 1; rr < KNN; ++rr) {
        float t2 = A[rr][c];
        for (int j = 0; j < c; ++j) t2 -= L[rr][j] * L[c][j];
        L[rr][c] = t2 / dc;
      }
    }
    float y[KNN], ww[KNN];
    for (int rr = 0; rr < KNN; ++rr) {
      float s = 1.0f;
      for (int j = 0; j < rr; ++j) s -= L[rr][j] * y[j];
      y[rr] = s / L[rr][rr];
    }
    for (int rr = KNN - 1; rr >= 0; --rr) {
      float s = y[rr];
      for (int j = rr + 1; j < KNN; ++j) s -= L[j][rr] * ww[j];
      ww[rr] = s / L[rr][rr];
    }
    float sum = 0.f;
    for (int t = 0; t < KNN; ++t) sum += ww[t];
    float inv = 1.0f / sum;
    for (int t = 0; t < KNN; ++t) w[row * KNN + t] = ww[t] * inv;
  }
}

// ---------------- utility zero / scatter M / Gershgorin ----------------
__global__ void k_zero(float* p, size_t n) {
  size_t i = blockIdx.x * (size_t)blockDim.x + threadIdx.x;
  size_t stride = gridDim.x * (size_t)blockDim.x;
  for (; i < n; i += stride) p[i] = 0.f;
}
__global__ void k_set0u(unsigned* p) { if (threadIdx.x == 0 && blockIdx.x == 0) *p = 0u; }

__global__ void k_scatter(const int* __restrict__ idx, const float* __restrict__ w,
                          float* __restrict__ M) {
  int i = blockIdx.x * blockDim.x + threadIdx.x;
  if (i >= N) return;
  int   r[KNN + 1]; float v[KNN + 1];
  r[0] = i; v[0] = 1.0f;
  for (int t = 0; t < KNN; ++t) { r[t + 1] = idx[i * KNN + t]; v[t + 1] = -w[i * KNN + t]; }
  for (int a = 0; a <= KNN; ++a)
    for (int b = 0; b <= KNN; ++b)
      atomicAdd(&M[(size_t)r[a] * N + r[b]], v[a] * v[b]);
}

__global__ void k_gersh(const float* __restrict__ M, unsigned* smax) {
  int row = blockIdx.x * 8 + (threadIdx.x >> 5);
  int lane = threadIdx.x & 31;
  float s = 0.f;
  const float* r = M + (size_t)row * N;
  for (int j = lane; j < N; j += 32) s += fabsf(r[j]);
  s = wave_sum(s);
  if (lane == 0) atomicMax(smax, __float_as_uint(s));  // positive floats order as uints
}

// ---------------- init subspace ----------------
__global__ void k_initV(float* __restrict__ V) {
  int i = blockIdx.x * blockDim.x + threadIdx.x;
  if (i >= N) return;
  V[(size_t)i * P + 0] = 1.0f;                           // trivial constant vector
  for (int p = 1; p < P; ++p) {
    unsigned h = (unsigned)i * 2654435761u ^ (unsigned)p * 2246822519u;
    h ^= h >> 15; h *= 2246822519u; h ^= h >> 13; h *= 3266489917u; h ^= h >> 16;
    V[(size_t)i * P + p] = ((h >> 8) * (1.0f / 8388608.0f)) - 1.0f;
  }
}

// ---- async (or fallback) staging of one 128-row panel of V into LDS ----
static __device__ __forceinline__ void stage_panel(const float* __restrict__ Vg,
                                                   int k0, float* dst) {
#ifdef USE_ASYNC_LDS
  const float* gp0 = Vg + (size_t)k0 * P;
#pragma unroll
  for (int t = 0; t < 6; ++t) {                         // 1536 b128 chunks / 256 thr
    int c = (threadIdx.x + t * 256) * 4;                // float offset, 16B aligned
    __builtin_amdgcn_global_load_async_to_lds_b128(
        (__attribute__((address_space(1))) b128_t*)(unsigned long long)(gp0 + c),
        (__attribute__((address_space(3))) b128_t*)(unsigned long long)(dst + c),
        0, 0);
  }
#else
  for (int t = threadIdx.x; t < 128 * P; t += blockDim.x)
    dst[t] = Vg[(size_t)k0 * P + t];
#endif
}

static __device__ __forceinline__ void wait_panel(int keepNext) {
#ifdef USE_ASYNC_LDS
  if (keepNext) __builtin_amdgcn_s_wait_asynccnt(6);    // keep next panel in flight
  else          __builtin_amdgcn_s_wait_asynccnt(0);
#else
  (void)keepNext;
#endif
}

// ---------------- Y = (useShift ? s*V - M V : M V)  (f32 WMMA + async LDS) ----
__global__ void k_mv(const float* __restrict__ M, const float* __restrict__ V,
                     float* __restrict__ Y, const unsigned* __restrict__ smax,
                     int useShift) {
  __shared__ __align__(16) float Vs[2][128 * P];        // 2 x 24 KB panels
  const int lane = threadIdx.x & 31;
  const int wave = threadIdx.x >> 5;
  const int rowBase = blockIdx.x * 128 + wave * 16;
  const int m  = lane & 15;
  const int kb = (lane >> 4) * 2;

  v8f accA[3], accB[3];
  v8f z = {0.f,0.f,0.f,0.f,0.f,0.f,0.f,0.f};
  for (int t = 0; t < 3; ++t) { accA[t] = z; accB[t] = z; }

  stage_panel(V, 0, Vs[0]);
  int buf = 0;
  for (int k0 = 0; k0 < N; k0 += 128) {
    int hasNext = (k0 + 128 < N);
    if (hasNext) stage_panel(V, k0 + 128, Vs[buf ^ 1]);
    wait_panel(hasNext);
    __syncthreads();
    const float* Vp = Vs[buf];
    const float* mrow = M + (size_t)(rowBase + m) * N + k0 + kb;
    for (int kk = 0; kk < 128; kk += 8) {
      __builtin_prefetch(mrow + kk + 512, 0, 1);
      v2f a0 = *(const v2f*)(mrow + kk);
      v2f a1 = *(const v2f*)(mrow + kk + 4);
#pragma unroll
      for (int t = 0; t < 3; ++t) {
        int nn = t * 16 + m;
        v2f b0, b1;
        b0.x = Vp[(kk + kb) * P + nn];
        b0.y = Vp[(kk + kb + 1) * P + nn];
        b1.x = Vp[(kk + 4 + kb) * P + nn];
        b1.y = Vp[(kk + 4 + kb + 1) * P + nn];
        accA[t] = wmma4(a0, b0, accA[t]);
        accB[t] = wmma4(a1, b1, accB[t]);
      }
    }
    __syncthreads();
    buf ^= 1;
  }
  float shift = __uint_as_float(*smax) * 1.0002f + 1.0f;  // > lambda_max(M)
  const int mb = rowBase + (lane >> 4) * 8;
#pragma unroll
  for (int t = 0; t < 3; ++t) {
    v8f acc = accA[t] + accB[t];
    int nn = t * 16 + m;
#pragma unroll
    for (int r = 0; r < 8; ++r) {
      float val = acc[r];
      if (useShift) val = shift * V[(size_t)(mb + r) * P + nn] - val;
      Y[(size_t)(mb + r) * P + nn] = val;
    }
  }
}

// ---- Spart[b] = A^T B over K-slice b (P x P, f32 WMMA, KSPL blocks x 9 waves) --
__global__ void k_gram(const float* __restrict__ A, const float* __restrict__ B,
                       float* __restrict__ Spart) {
  const int lane = threadIdx.x & 31;
  const int wave = threadIdx.x >> 5;   // 0..8
  const int tm = wave / 3, tn = wave % 3;
  const int m  = lane & 15;
  const int kb = (lane >> 4) * 2;
  const int kBeg = blockIdx.x * (N / KSPL);
  const int kEnd = kBeg + (N / KSPL);
  v8f acc[4];
  v8f z = {0.f,0.f,0.f,0.f,0.f,0.f,0.f,0.f};
  for (int u = 0; u < 4; ++u) acc[u] = z;
  for (int k0 = kBeg; k0 < kEnd; k0 += 16) {
#pragma unroll
    for (int u = 0; u < 4; ++u) {
      int k = k0 + u * 4;
      v2f a, b;
      a.x = A[(size_t)(k + kb) * P + tm * 16 + m];
      a.y = A[(size_t)(k + kb + 1) * P + tm * 16 + m];
      b.x = B[(size_t)(k + kb) * P + tn * 16 + m];
      b.y = B[(size_t)(k + kb + 1) * P + tn * 16 + m];
      acc[u] = wmma4(a, b, acc[u]);
    }
  }
  v8f s = (acc[0] + acc[1]) + (acc[2] + acc[3]);
  const int mb = tm * 16 + (lane >> 4) * 8;
  float* Sp = Spart + (size_t)blockIdx.x * (P * P);
#pragma unroll
  for (int r = 0; r < 8; ++r) Sp[(mb + r) * P + tn * 16 + m] = s[r];
}

// deterministic fixed-order reduction of the K-split partials
__global__ void k_sumparts(const float* __restrict__ Spart, float* __restrict__ S) {
  int t = blockIdx.x * blockDim.x + threadIdx.x;
  if (t >= P * P) return;
  float s = 0.f;
  for (int b = 0; b < KSPL; ++b) s += Spart[(size_t)b * (P * P) + t];
  S[t] = s;
}

// ---------------- Cholesky S = R^T R (upper R), lane-parallel ----------------
__global__ void k_chol(const float* __restrict__ S, float* __restrict__ R) {
  __shared__ float A[P][P];
  __shared__ float Rl[P][P];
  int lane = threadIdx.x;  // blockDim 32
  for (int q = lane; q < P * P; q += 32) { A[q / P][q % P] = S[q]; Rl[q / P][q % P] = 0.f; }
  __syncthreads();
  if (lane == 0) {
    float tr = 0.f;
    for (int i = 0; i < P; ++i) tr += A[i][i];
    float jit = 1e-7f * tr + 1e-20f;
    for (int i = 0; i < P; ++i) A[i][i] += jit;
  }
  __syncthreads();
  for (int c = 0; c < P; ++c) {
    if (lane == 0) {
      float s = A[c][c];
      for (int j = 0; j < c; ++j) s -= Rl[j][c] * Rl[j][c];
      Rl[c][c] = sqrtf(fmaxf(s, 1e-20f));
    }
    __syncthreads();
    float dc = Rl[c][c];
    for (int n = c + 1 + lane; n < P; n += 32) {
      float t = A[c][n];
      for (int j = 0; j < c; ++j) t -= Rl[j][c] * Rl[j][n];
      Rl[c][n] = t / dc;
    }
    __syncthreads();
  }
  for (int q = lane; q < P * P; q += 32) R[q] = Rl[q / P][q % P];
}

// ---------------- Vout = Yin * R^{-1} (per-row forward substitution) ----------
__global__ void k_trsm(const float* __restrict__ Yin, const float* __restrict__ R,
                       float* __restrict__ Vout) {
  __shared__ float Rs[P * P];
  for (int t = threadIdx.x; t < P * P; t += blockDim.x) Rs[t] = R[t];
  __syncthreads();
  int i = blockIdx.x * blockDim.x + threadIdx.x;
  if (i >= N) return;
  float v[P];
  for (int c = 0; c < P; ++c) {
    float s = Yin[(size_t)i * P + c];
    for (int j = 0; j < c; ++j) s -= v[j] * Rs[j * P + c];
    v[c] = s / Rs[c * P + c];
  }
  for (int c = 0; c < P; ++c) Vout[(size_t)i * P + c] = v[c];
}

// ---------------- 48x48 Jacobi eigensolver (lane-parallel) + ascending sort ----
__global__ void k_jacobi(const float* __restrict__ Hg, float* __restrict__ Ug) {
  __shared__ float h[P][P];
  __shared__ float u[P][P];
  int lane = threadIdx.x;  // blockDim 32
  for (int q = lane; q < P * P; q += 32) {
    int i = q / P, j = q % P;
    h[i][j] = 0.5f * (Hg[q] + Hg[j * P + i]);   // symmetrize
    u[i][j] = (i == j) ? 1.f : 0.f;
  }
  __syncthreads();
  for (int sweep = 0; sweep < 16; ++sweep)
    for (int p = 0; p < P - 1; ++p)
      for (int q = p + 1; q < P; ++q) {
        float apq = h[p][q];                    // uniform across lanes
        float hpp = h[p][p], hqq = h[q][q];
        if (fabsf(apq) > 1e-12f) {              // uniform branch
          float theta = (hqq - hpp) / (2.f * apq);
          float tt = (theta >= 0.f ? 1.f : -1.f) /
                     (fabsf(theta) + sqrtf(theta * theta + 1.f));
          float c = 1.f / sqrtf(tt * tt + 1.f), s = tt * c;
          for (int k = lane; k < P; k += 32) {  // column rotation + eigvecs
            float hkp = h[k][p], hkq = h[k][q];
            h[k][p] = c * hkp - s * hkq; h[k][q] = s * hkp + c * hkq;
            float ukp = u[k][p], ukq = u[k][q];
            u[k][p] = c * ukp - s * ukq; u[k][q] = s * ukp + c * ukq;
          }
          __syncthreads();
          for (int k = lane; k < P; k += 32) {  // row rotation
            float hpk = h[p][k], hqk = h[q][k];
            h[p][k] = c * hpk - s * hqk; h[q][k] = s * hpk + c * hqk;
          }
        }
        __syncthreads();
      }
  if (lane == 0) {
    for (int a = 0; a < P - 1; ++a) {           // sort eigenvalues ascending
      int best = a;
      for (int b = a + 1; b < P; ++b)
        if (h[b][b] < h[best][best]) best = b;
      if (best != a) {
        float tmp = h[a][a]; h[a][a] = h[best][best]; h[best][best] = tmp;
        for (int k = 0; k < P; ++k) {
          float tu = u[k][a]; u[k][a] = u[k][best]; u[k][best] = tu;
        }
      }
    }
  }
  __syncthreads();
  for (int q = lane; q < P * P; q += 32) Ug[q] = u[q / P][q % P];
}

// ---------------- out = V * U[:, 1:33] ----------------
__global__ void k_project(const float* __restrict__ V, const float* __restrict__ U,
                          float* __restrict__ out) {
  __shared__ float Us[P * P];
  for (int t = threadIdx.x; t < P * P; t += blockDim.x) Us[t] = U[t];
  __syncthreads();
  int i = blockIdx.x * blockDim.x + threadIdx.x;
  if (i >= N) return;
  float v[P];
  for (int p = 0; p < P; ++p) v[p] = V[(size_t)i * P + p];
  for (int c = 0; c < NC; ++c) {
    float s = 0.f;
    for (int p = 0; p < P; ++p) s += v[p] * Us[p * P + (c + 1)];  // skip trivial
    out[(size_t)i * NC + c] = s;
  }
}

extern "C" void kernel_launch(void* const* d_in, const int* in_sizes, int n_in,
                              void* d_out, int out_size, void* d_ws, size_t ws_size,
                              hipStream_t stream) {
  (void)in_sizes; (void)n_in; (void)out_size; (void)ws_size;
  const float* x = (const float*)d_in[0];
  float* out = (float*)d_out;

  char* base = (char*)d_ws;
  size_t off = 0;
  auto alloc = [&](size_t bytes) {
    void* p = base + off;
    off += (bytes + 255) & ~(size_t)255;
    return p;
  };
  float*    M     = (float*)alloc((size_t)N * N * sizeof(float));  // d2, then M
  float*    sq    = (float*)alloc((size_t)N * sizeof(float));
  int*      idx   = (int*)  alloc((size_t)N * KNN * sizeof(int));
  float*    w     = (float*)alloc((size_t)N * KNN * sizeof(float));
  float*    Va    = (float*)alloc((size_t)N * P * sizeof(float));
  float*    Vb    = (float*)alloc((size_t)N * P * sizeof(float));
  float*    Spart = (float*)alloc((size_t)KSPL * P * P * sizeof(float));
  float*    S     = (float*)alloc((size_t)P * P * sizeof(float));
  float*    R     = (float*)alloc((size_t)P * P * sizeof(float));
  float*    H     = (float*)alloc((size_t)P * P * sizeof(float));
  float*    U     = (float*)alloc((size_t)P * P * sizeof(float));
  unsigned* smax  = (unsigned*)alloc(sizeof(unsigned));

  // 1. distances + kNN
  k_rownorm<<<N / 8, 256, 0, stream>>>(x, sq);
  k_dist<<<dim3(N / 64, N / 128), 256, 0, stream>>>(x, sq, M);   // M holds d2
  k_topk<<<N, 32, 0, stream>>>(M, idx);

  // 2. barycentric weights
  k_weights<<<N, 32, 0, stream>>>(x, idx, w);

  // 3. M = (I-W)^T (I-W)
  k_zero<<<1024, 256, 0, stream>>>(M, (size_t)N * N);
  k_scatter<<<N / 256, 256, 0, stream>>>(idx, w, M);
  k_set0u<<<1, 32, 0, stream>>>(smax);
  k_gersh<<<N / 8, 256, 0, stream>>>(M, smax);

  // 4. shifted subspace iteration for bottom eigenvectors
  k_initV<<<N / 256, 256, 0, stream>>>(Va);
  k_gram<<<KSPL, 288, 0, stream>>>(Va, Va, Spart);
  k_sumparts<<<(P * P + 255) / 256, 256, 0, stream>>>(Spart, S);
  k_chol<<<1, 32, 0, stream>>>(S, R);
  k_trsm<<<N / 256, 256, 0, stream>>>(Va, R, Vb);
  { float* t = Va; Va = Vb; Vb = t; }
  for (int it = 0; it < ITERS; ++it) {
    k_mv<<<N / 128, 256, 0, stream>>>(M, Va, Vb, smax, 1);  // Vb = sVa - M Va
    k_gram<<<KSPL, 288, 0, stream>>>(Vb, Vb, Spart);
    k_sumparts<<<(P * P + 255) / 256, 256, 0, stream>>>(Spart, S);
    k_chol<<<1, 32, 0, stream>>>(S, R);
    k_trsm<<<N / 256, 256, 0, stream>>>(Vb, R, Va);         // Va = orth(Vb)
  }

  // 5. Rayleigh-Ritz + projection to 32 components (skip trivial)
  k_mv<<<N / 128, 256, 0, stream>>>(M, Va, Vb, smax, 0);    // Vb = M Va
  k_gram<<<KSPL, 288, 0, stream>>>(Va, Vb, Spart);
  k_sumparts<<<(P * P + 255) / 256, 256, 0, stream>>>(Spart, H);
  k_jacobi<<<1, 32, 0, stream>>>(H, U);
  k_project<<<N / 256, 256, 0, stream>>>(Va, U, out);
}